// HLR_15264313770621
// MI455X (gfx1250) — compile-verified
//
#include <hip/hip_runtime.h>
#include <cstdint>
#include <cstddef>

// ---------------------------------------------------------------------------
// Half-life regression forward pass, MI455X (gfx1250).
// B=256, N=512, T=200.
// Memory-bound (~522 MB @ 23.3 TB/s ~ 22us). Strategy:
//   * per-(b,n) row: async-DMA 600 floats of stats into LDS
//     (global_load_async_to_lds_b128, ASYNCcnt) -> consume from LDS.
//   * non-temporal stores for all streamed outputs.
//   * native v_exp_f32 / v_rcp_f32 for exp2/sigmoid.
// ---------------------------------------------------------------------------

#define B_DIM 256
#define N_DIM 512
#define T_DIM 200
#define TM1   199

#define INV_SECONDS_PER_DAY (1.0f / 86400.0f)
#define MIN_HL 0.010416666666666666f   /* 15 / 1440 */
#define MAX_HL 274.0f
#define LOG2E  1.4426950408889634f

__device__ __forceinline__ float fast_exp2(float x) {
  return __builtin_amdgcn_exp2f(x);           // v_exp_f32
}
__device__ __forceinline__ float fast_rcp(float x) {
  return __builtin_amdgcn_rcpf(x);            // v_rcp_f32 (1 ulp)
}

__device__ __forceinline__ float half_life_of(float s0, float s1, float s2,
                                              float th0, float th1, float th2) {
  float logit = fmaf(s0, th0, fmaf(s1, th1, s2 * th2));
  float hl = fast_exp2(logit);
  return fminf(fmaxf(hl, MIN_HL), MAX_HL);
}

__device__ __forceinline__ float recall_p(float dt_days, float hl) {
  // sigmoid(exp2(-dt/hl)) with native exp2/rcp
  float y = fast_exp2(-dt_days * fast_rcp(hl));
  return fast_rcp(1.0f + fast_exp2(-y * LOG2E));
}

// One block per (b,n) row. Threads 0..198 produce p/half_life for t=tid,
// thread 199 copies x0 into x_pred[...,0].
__global__ __launch_bounds__(256)
void hlr_main_kernel(const float* __restrict__ stats,   // (B,N,T,3)
                     const float* __restrict__ tarr,    // (B,T)
                     const float* __restrict__ x0,      // (B,N)
                     const float* __restrict__ theta,   // (3,)
                     float* __restrict__ out_xpred,     // (B,N,T)
                     float* __restrict__ out_hl)        // (B,N,T-1)
{
  __shared__ __align__(16) float srow[T_DIM * 3];       // 600 floats = 2400 B

  const int row = blockIdx.x;            // b*N + n  in [0, 131072)
  const int tid = threadIdx.x;
  const int b   = row >> 9;              // row / N_DIM

  // ---- async bulk copy: this row's 600 stats floats -> LDS ---------------
  // 2400 B = 150 lanes x b128; row base is 2400B-strided from a 256B-aligned
  // base, so every 16B chunk is naturally aligned.
  if (tid < 150) {
    unsigned lds_addr =
        (unsigned)(uintptr_t)(const void*)&srow[tid * 4];
    unsigned long long gaddr =
        (unsigned long long)(uintptr_t)(stats + (size_t)row * (T_DIM * 3) + tid * 4);
    asm volatile("global_load_async_to_lds_b128 %0, %1, off"
                 :: "v"(lds_addr), "v"(gaddr)
                 : "memory");
  }
  asm volatile("s_wait_asynccnt 0" ::: "memory");   // issuing waves drain DMA
  __syncthreads();                                  // publish LDS to all waves

  const float th0 = theta[0];
  const float th1 = theta[1];
  const float th2 = theta[2];

  if (tid < TM1) {
    const float s0 = srow[tid * 3 + 0];
    const float s1 = srow[tid * 3 + 1];
    const float s2 = srow[tid * 3 + 2];

    const float hl = half_life_of(s0, s1, s2, th0, th1, th2);

    const float t0 = tarr[b * T_DIM + tid];
    const float t1 = tarr[b * T_DIM + tid + 1];
    const float dt = (t1 - t0) * INV_SECONDS_PER_DAY;

    const float p = recall_p(dt, hl);

    __builtin_nontemporal_store(p,  &out_xpred[(size_t)row * T_DIM + tid + 1]);
    __builtin_nontemporal_store(hl, &out_hl[(size_t)row * TM1 + tid]);
  } else if (tid == TM1) {
    __builtin_nontemporal_store(x0[row], &out_xpred[(size_t)row * T_DIM]);
  }
}

// x_item_pred[b,t] = sigmoid(exp2(-dt[b,t]/hl(stats[b, items[b,t+1], t, :])))
// Tiny gather kernel: 256*199 = 50,944 elements; independent of hlr_main.
__global__ __launch_bounds__(256)
void hlr_item_kernel(const float* __restrict__ stats,   // (B,N,T,3)
                     const float* __restrict__ tarr,    // (B,T)
                     const int*   __restrict__ items,   // (B,T)
                     const float* __restrict__ theta,   // (3,)
                     float* __restrict__ out_item)      // (B,T-1)
{
  const int idx = blockIdx.x * blockDim.x + threadIdx.x;
  if (idx >= B_DIM * TM1) return;

  const int b = idx / TM1;
  const int t = idx - b * TM1;

  const int item = items[b * T_DIM + t + 1];
  const size_t s = ((size_t)(b * N_DIM + item) * T_DIM + t) * 3;

  const float s0 = stats[s + 0];
  const float s1 = stats[s + 1];
  const float s2 = stats[s + 2];

  const float hl = half_life_of(s0, s1, s2, theta[0], theta[1], theta[2]);

  const float t0 = tarr[b * T_DIM + t];
  const float t1 = tarr[b * T_DIM + t + 1];
  const float dt = (t1 - t0) * INV_SECONDS_PER_DAY;

  out_item[idx] = recall_p(dt, hl);
}

extern "C" void kernel_launch(void* const* d_in, const int* in_sizes, int n_in,
                              void* d_out, int out_size, void* d_ws, size_t ws_size,
                              hipStream_t stream) {
  (void)in_sizes; (void)n_in; (void)out_size; (void)d_ws; (void)ws_size;

  // setup_inputs() order: x0, t, items, stats, theta
  const float* x0    = (const float*)d_in[0];
  const float* tarr  = (const float*)d_in[1];
  const int*   items = (const int*)  d_in[2];
  const float* stats = (const float*)d_in[3];
  const float* theta = (const float*)d_in[4];

  float* out = (float*)d_out;
  // Tuple return order, flattened: x_pred, x_item_pred, half_life
  float* out_xpred = out;                                           // B*N*T
  float* out_item  = out_xpred + (size_t)B_DIM * N_DIM * T_DIM;     // B*(T-1)
  float* out_hl    = out_item  + (size_t)B_DIM * TM1;               // B*N*(T-1)

  hlr_main_kernel<<<B_DIM * N_DIM, 256, 0, stream>>>(
      stats, tarr, x0, theta, out_xpred, out_hl);

  const int n_item = B_DIM * TM1;
  hlr_item_kernel<<<(n_item + 255) / 256, 256, 0, stream>>>(
      stats, tarr, items, theta, out_item);
}